// LCNN_29326036697934
// MI455X (gfx1250) — compile-verified
//
#include <hip/hip_runtime.h>

// Gauge-equivariant lattice conv for B=2, L=16, D=4, K=2, N_in=N_out=8, Nc=3.
//   out[s,i,a,b] = sum_{mu,k,j,c,d} omega[i,j,mu,k] * U[s,mu,k,a,c] * W[s+k*e_mu, j,c,d] * U[s,mu,k,b,d]
// Strategy per wave (one x4-line of 16 sites):
//   1) channel mix Wmix[i, site, cd] via v_wmma_f32_16x16x4_f32 (M=i, N=site, K=j in 2 chained WMMAs)
//   2) ds_bpermute spreads i=4..7 to lanes 16..31 so all 32 lanes sandwich U*Wmix*U^T on VALU
// fp32 end-to-end (exact), L2-resident re-reads of W.

typedef __attribute__((ext_vector_type(2))) float v2f;
typedef __attribute__((ext_vector_type(8))) float v8f;

__global__ __launch_bounds__(256)
void lcnn_wmma_kernel(const float* __restrict__ W,
                      const float* __restrict__ U,
                      const float* __restrict__ Om,
                      float* __restrict__ Out)
{
    const int tid  = threadIdx.x;
    const int wave = tid >> 5;
    const int lane = tid & 31;
    const int line = blockIdx.x * 8 + wave;       // 0..8191  = ((b*16+x1)*16+x2)*16+x3

    const int s     = lane & 15;                  // site within line / N column / src lane
    const int hh    = lane >> 4;                  // lane-half
    const int jadd  = hh * 2;                     // K-slot offset of this half in A/B layout
    const int ibase = hh * 4;                     // output channels this lane sandwiches

    const int x3 = line & 15;
    const int x2 = (line >> 4) & 15;
    const int x1 = (line >> 8) & 15;

    const int mySite = line * 16 + s;             // this lane's (output) site

    float outAcc[36];                             // [ii(4)][ab(9)], i = ibase+ii
    #pragma unroll
    for (int q = 0; q < 36; ++q) outAcc[q] = 0.f;

    #pragma unroll 1
    for (int t = 0; t < 20; ++t) {
        const int mu = t & 3;
        const int ik = t >> 2;
        const int k  = ik - 2;

        // ---- shifted W base: W'[x] = W[(x+k) mod 16] along axis mu ----
        int wline  = line;
        int sshift = s;
        if      (mu == 0) wline = (line & ~(15 << 8)) | (((x1 + k) & 15) << 8);
        else if (mu == 1) wline = (line & ~(15 << 4)) | (((x2 + k) & 15) << 4);
        else if (mu == 2) wline = (line & ~15)        |  ((x3 + k) & 15);
        else              sshift = (s + k + 16) & 15;

        // per-lane W' base (floats); fold in this lane-half's K-slot j offset (jadd*9)
        const int wAddr = wline * 1152 + sshift * 72 + jadd * 9;

        // ---- A operand: omega[i=s, j, mu, ik]; rows 8..15 are zero ----
        float a0v0 = 0.f, a0v1 = 0.f, a1v0 = 0.f, a1v1 = 0.f;
        if (s < 8) {
            const int ob = s * 160 + mu * 5 + ik;            // + j*20
            a0v0 = Om[ob + (jadd + 0) * 20];                 // h=0: K = jadd, jadd+1
            a0v1 = Om[ob + (jadd + 1) * 20];
            a1v0 = Om[ob + (jadd + 4) * 20];                 // h=1: j = 4 + ...
            a1v1 = Om[ob + (jadd + 5) * 20];
        }
        const v2f A0 = {a0v0, a0v1};
        const v2f A1 = {a1v0, a1v1};

        // ---- U at the (unshifted) output site ----
        const float* up = U + mySite * 180 + (mu * 5 + ik) * 9;
        float u[9];
        #pragma unroll
        for (int q = 0; q < 9; ++q) u[q] = up[q];

        // ---- channel mixing on the matrix pipe: Wmix[i, site, cd] ----
        v8f acc[9];
        #pragma unroll
        for (int cd = 0; cd < 9; ++cd) {
            v8f c = {0.f, 0.f, 0.f, 0.f, 0.f, 0.f, 0.f, 0.f};
            const v2f B0 = { W[wAddr +      cd], W[wAddr +  9 + cd] };   // j = 0..3 half
            const v2f B1 = { W[wAddr + 36 + cd], W[wAddr + 45 + cd] };   // j = 4..7 half
            c = __builtin_amdgcn_wmma_f32_16x16x4_f32(false, A0, false, B0, (short)0, c, false, false);
            c = __builtin_amdgcn_wmma_f32_16x16x4_f32(false, A1, false, B1, (short)0, c, false, false);
            acc[cd] = c;
        }

        // ---- redistribute rows 4..7 to the upper lane-half (ds_bpermute from lane s) ----
        float w[9][4];
        const int pidx = s << 2;
        #pragma unroll
        for (int cd = 0; cd < 9; ++cd) {
            #pragma unroll
            for (int ii = 0; ii < 4; ++ii) {
                const float hi = __int_as_float(
                    __builtin_amdgcn_ds_bpermute(pidx, __float_as_int(acc[cd][4 + ii])));
                w[cd][ii] = (hh == 0) ? acc[cd][ii] : hi;
            }
        }

        // ---- sandwich: out_i += U * Wmix_i * U^T  (i = ibase+ii) ----
        #pragma unroll
        for (int ii = 0; ii < 4; ++ii) {
            float T[9];                                        // T[c,b] = sum_d Wmix[c,d] * U[b,d]
            #pragma unroll
            for (int c = 0; c < 3; ++c)
                #pragma unroll
                for (int b = 0; b < 3; ++b)
                    T[c*3+b] = w[c*3+0][ii]*u[b*3+0]
                             + w[c*3+1][ii]*u[b*3+1]
                             + w[c*3+2][ii]*u[b*3+2];
            #pragma unroll
            for (int a = 0; a < 3; ++a)
                #pragma unroll
                for (int b = 0; b < 3; ++b)
                    outAcc[ii*9 + a*3 + b] += u[a*3+0]*T[0+b]
                                            + u[a*3+1]*T[3+b]
                                            + u[a*3+2]*T[6+b];
        }
    }

    // ---- store: 36 contiguous floats per lane, 16B aligned ----
    float4* op = (float4*)(Out + mySite * 72 + ibase * 9);
    #pragma unroll
    for (int q = 0; q < 9; ++q)
        op[q] = make_float4(outAcc[q*4+0], outAcc[q*4+1], outAcc[q*4+2], outAcc[q*4+3]);
}

extern "C" void kernel_launch(void* const* d_in, const int* in_sizes, int n_in,
                              void* d_out, int out_size, void* d_ws, size_t ws_size,
                              hipStream_t stream) {
    const float* W  = (const float*)d_in[0];   // (2,16,16,16,16,8,3,3)
    const float* U  = (const float*)d_in[1];   // (2,16,16,16,16,4,5,3,3)
    const float* Om = (const float*)d_in[2];   // (8,8,4,5)
    float* Out = (float*)d_out;                // (2,16,16,16,16,8,3,3)
    (void)in_sizes; (void)n_in; (void)out_size; (void)d_ws; (void)ws_size;

    // 8192 x4-lines, one wave each: 1024 blocks x 256 threads (8 waves/block)
    lcnn_wmma_kernel<<<dim3(1024), dim3(256), 0, stream>>>(W, U, Om, Out);
}